// GNN_LRMC_16234976379023
// MI455X (gfx1250) — compile-verified
//
#include <hip/hip_runtime.h>

typedef __attribute__((ext_vector_type(2))) float v2f;
typedef __attribute__((ext_vector_type(4))) float v4f;
typedef __attribute__((ext_vector_type(8))) float v8f;

#define NROWS  100000
#define MCOLS  50000
#define NNODES 150000   // NROWS + MCOLS, divisible by 16 (9375 row tiles)
#define DIM    32
#define NEDGE  5000000

// h = x @ W^T + b ; agg = h (self-loop contribution folded in).
// One wave computes a 16-row x 32-col slab via V_WMMA_F32_16X16X4_F32.
__global__ void gnn_gemm_selfloop(const float* __restrict__ x,
                                  const float* __restrict__ W,
                                  const float* __restrict__ bias,
                                  float* __restrict__ h,
                                  float* __restrict__ agg)
{
    const int wavesPerBlock = blockDim.x >> 5;
    const int wave = blockIdx.x * wavesPerBlock + (threadIdx.x >> 5);
    const int numRowTiles = NNODES / 16;
    if (wave >= numRowTiles) return;          // wave-uniform: EXEC stays all-ones

    const int lane = threadIdx.x & 31;
    const int half = lane >> 4;               // K-half select (ISA A/B layout)
    const int lm   = lane & 15;               // M for A, N for B/C/D
    const int rowBase = wave * 16;

    const float* xrow = x + (size_t)(rowBase + lm) * DIM;

#pragma unroll
    for (int jt = 0; jt < 2; ++jt) {
        const int colBase = jt * 16;
        const float* wrow = W + (size_t)(colBase + lm) * DIM;  // B[k][n] = W[n][k]
        v8f c = {0.f, 0.f, 0.f, 0.f, 0.f, 0.f, 0.f, 0.f};
#pragma unroll
        for (int k0 = 0; k0 < DIM; k0 += 4) {
            const int ka = k0 + 2 * half;     // even -> 8B-aligned float2 loads
            v2f a = *(const v2f*)(xrow + ka); // A: M=lm, K=ka,ka+1
            v2f b = *(const v2f*)(wrow + ka); // B: N=lm, K=ka,ka+1
            c = __builtin_amdgcn_wmma_f32_16x16x4_f32(
                    /*neg_a=*/false, a, /*neg_b=*/false, b,
                    /*c_mod=*/(short)0, c, /*reuse_a=*/false, /*reuse_b=*/false);
        }
        const float bv = bias[colBase + lm];
#pragma unroll
        for (int v = 0; v < 8; ++v) {
            const int row = rowBase + v + 8 * half;           // D layout: M = v + 8*half
            const float val = c[v] + bv;
            const size_t idx = (size_t)row * DIM + colBase + lm;
            h[idx]   = val;
            agg[idx] = val;                                   // self-loop: agg starts at h
        }
    }
}

// One wave per edge: lanes 0..31 = the 32 feature dims. Coalesced 128B gather
// from h[src], 32 same-cacheline f32 atomics into agg[dst] (L2-resident).
__global__ void gnn_scatter(const float* __restrict__ h,
                            const int* __restrict__ edge,
                            float* __restrict__ agg)
{
    const long long g = (long long)blockIdx.x * blockDim.x + threadIdx.x;
    const long long total = (long long)NEDGE * 32;
    if (g >= total) return;
    const int e = (int)(g >> 5);
    const int d = (int)(g & 31);
    const int src = edge[e];
    const int dst = edge[NEDGE + e];
    const float val = h[(size_t)src * DIM + d];
    unsafeAtomicAdd(agg + (size_t)dst * DIM + d, val);        // global_atomic_add_f32 (no return)
}

// x = relu(agg), vectorized 16B.
__global__ void gnn_relu(const float* __restrict__ agg, float* __restrict__ out)
{
    const int i = blockIdx.x * blockDim.x + threadIdx.x;
    const int n4 = NNODES * DIM / 4;
    if (i >= n4) return;
    v4f v = ((const v4f*)agg)[i];
    v4f r;
    r.x = fmaxf(v.x, 0.f);
    r.y = fmaxf(v.y, 0.f);
    r.z = fmaxf(v.z, 0.f);
    r.w = fmaxf(v.w, 0.f);
    ((v4f*)out)[i] = r;
}

extern "C" void kernel_launch(void* const* d_in, const int* in_sizes, int n_in,
                              void* d_out, int out_size, void* d_ws, size_t ws_size,
                              hipStream_t stream)
{
    const float* row_embed = (const float*)d_in[0];
    const float* col_embed = (const float*)d_in[1];
    const float* Ws[3] = {(const float*)d_in[2], (const float*)d_in[4], (const float*)d_in[6]};
    const float* bs[3] = {(const float*)d_in[3], (const float*)d_in[5], (const float*)d_in[7]};
    const int*   edge  = (const int*)d_in[8];
    float* out = (float*)d_out;

    const size_t SZ = (size_t)NNODES * DIM;
    float* x   = (float*)d_ws;
    float* h   = x + SZ;
    float* agg = h + SZ;

    // x = concat(row_embed, col_embed)
    hipMemcpyAsync(x, row_embed, (size_t)NROWS * DIM * sizeof(float),
                   hipMemcpyDeviceToDevice, stream);
    hipMemcpyAsync(x + (size_t)NROWS * DIM, col_embed, (size_t)MCOLS * DIM * sizeof(float),
                   hipMemcpyDeviceToDevice, stream);

    const int gemmBlocks = (NNODES / 16 + 7) / 8;             // 8 waves / 256-thread block
    const long long scatterThreads = (long long)NEDGE * 32;
    const int scatBlocks = (int)((scatterThreads + 255) / 256);
    const int reluBlocks = (NNODES * DIM / 4 + 255) / 256;

    for (int l = 0; l < 3; ++l) {
        gnn_gemm_selfloop<<<gemmBlocks, 256, 0, stream>>>(x, Ws[l], bs[l], h, agg);
        gnn_scatter<<<scatBlocks, 256, 0, stream>>>(h, edge, agg);
        gnn_relu<<<reluBlocks, 256, 0, stream>>>(agg, (l == 2) ? out : x);
    }
}